// HungarianCoordsMatcher_61392262529165
// MI455X (gfx1250) — compile-verified
//
#include <hip/hip_runtime.h>

// ---------------------------------------------------------------------------
// Hungarian (DETR) matcher for MI455X / gfx1250.
//
// Stage 1: negated softmax probs            (B*Q rows of NC=20)      -> ws
// Stage 2: cost matrix via WMMA             C_T[b][t][q] (T x Q)     -> ws
//          class cost = onehot(labels) x (-prob)  with
//          V_WMMA_F32_16X16X4_F32, K = 20 classes = 5 chained WMMAs.
//          bbox L1 cost added in the accumulator epilogue.
// Stage 3: JV shortest-augmenting-path LSA, one workgroup per batch,
//          LDS-resident duals/path state, cost rows streamed from L2
//          (entire 32MB cost matrix fits the 192MB global L2).
// ---------------------------------------------------------------------------

#define BS 64
#define QN 1024
#define TN 128
#define DD 2
#define NC 20
#define INFF 1e30f

typedef __attribute__((ext_vector_type(2))) float v2f;
typedef __attribute__((ext_vector_type(8))) float v8f;

// ---------------------------------------------------------------- stage 1 --
__global__ void softmax_neg_kernel(const float* __restrict__ logits,
                                   float* __restrict__ negP) {
  int row = blockIdx.x * blockDim.x + threadIdx.x;   // b*Q + q
  if (row >= BS * QN) return;
  const float* lg = logits + (size_t)row * NC;
  float l[NC];
  float mx = -INFF;
#pragma unroll
  for (int c = 0; c < NC; ++c) { l[c] = lg[c]; mx = fmaxf(mx, l[c]); }
  float s = 0.f;
#pragma unroll
  for (int c = 0; c < NC; ++c) { l[c] = expf(l[c] - mx); s += l[c]; }
  float ninv = -1.0f / s;
  float* o = negP + (size_t)row * NC;
#pragma unroll
  for (int c = 0; c < NC; ++c) o[c] = l[c] * ninv;   // -softmax
}

// ---------------------------------------------------------------- stage 2 --
// One wave per 16(t) x 16(q) tile of C_T.  M=targets, N=queries, K=classes.
// A(16x4 f32): lanes 0-15 m=lane {K=kb+0,kb+1}; lanes 16-31 m=lane-16 {kb+2,kb+3}
// B(4x16 f32): lanes 0-15 n=lane rows {kb+0,kb+1}; lanes 16-31 rows {kb+2,kb+3}
// C/D(16x16 f32): VGPR r: lanes 0-15 -> (m=r, n=lane); lanes 16-31 -> (m=r+8).
__global__ void cost_wmma_kernel(const float* __restrict__ negP,
                                 const float* __restrict__ pred_coords,
                                 const int*   __restrict__ tgt_labels,
                                 const float* __restrict__ tgt_joints,
                                 float* __restrict__ cost) {
  int wave = (int)((blockIdx.x * blockDim.x + threadIdx.x) >> 5);
  int lane = (int)(threadIdx.x & 31);
  int qt = wave & 63;          // Q/16 = 64 tiles
  int tt = (wave >> 6) & 7;    // T/16 = 8 tiles
  int b  = wave >> 9;          // batch
  int q0 = qt << 4, t0 = tt << 4;
  int hi = lane >> 4;          // 0: lanes 0-15, 1: lanes 16-31
  int ln = lane & 15;

  int labelA = tgt_labels[b * TN + t0 + ln];              // A-matrix row m=ln
  const float* Pq = negP + (size_t)(b * QN + q0 + ln) * NC; // B-matrix col n=ln

  v8f acc{};
#pragma unroll
  for (int kb = 0; kb < NC; kb += 4) {
    int kA = kb + hi * 2;
    v2f a, bm;
    a.x  = (labelA == kA)     ? 1.0f : 0.0f;   // one-hot rows (exact)
    a.y  = (labelA == kA + 1) ? 1.0f : 0.0f;
    bm.x = Pq[kA];                              // -prob, f32 exact
    bm.y = Pq[kA + 1];
    acc = __builtin_amdgcn_wmma_f32_16x16x4_f32(
        /*neg_a=*/false, a, /*neg_b=*/false, bm,
        /*c_mod=*/(short)0, acc, /*reuse_a=*/false, /*reuse_b=*/false);
  }

  // Epilogue: add L1 bbox cost (D=2) and store C_T[b][t][q].
  int q = q0 + ln;
  float px = pred_coords[(size_t)(b * QN + q) * DD + 0];
  float py = pred_coords[(size_t)(b * QN + q) * DD + 1];
#pragma unroll
  for (int r = 0; r < 8; ++r) {
    int t = t0 + r + hi * 8;
    float tx = tgt_joints[(size_t)(b * TN + t) * DD + 0];
    float ty = tgt_joints[(size_t)(b * TN + t) * DD + 1];
    cost[((size_t)b * TN + t) * QN + q] =
        fabsf(px - tx) + fabsf(py - ty) + acc[r];
  }
}

// ---------------------------------------------------------------- stage 3 --
// JV / shortest augmenting path, rows = targets (128), cols = queries (1024).
// One block (256 threads = 8 waves) per batch; all solver state in LDS.
__global__ void lsa_kernel(const float* __restrict__ cost,
                           int* __restrict__ pred_out,
                           int* __restrict__ tgt_out) {
  __shared__ float v_[QN];
  __shared__ float spc[QN];
  __shared__ int   path[QN];
  __shared__ int   SC[QN];
  __shared__ int   row4col[QN];
  __shared__ float u_[TN];
  __shared__ int   col4row[TN];
  __shared__ float redV[8];
  __shared__ int   redJ[8];
  __shared__ float s_minVal;
  __shared__ int   s_i, s_sink;

  const int b   = blockIdx.x;
  const int tid = (int)threadIdx.x;
  const float* C = cost + (size_t)b * TN * QN;

  for (int j = tid; j < QN; j += 256) { v_[j] = 0.f; row4col[j] = -1; }
  if (tid < TN) { u_[tid] = 0.f; col4row[tid] = -1; }
  __syncthreads();

  for (int cur = 0; cur < TN; ++cur) {
    for (int j = tid; j < QN; j += 256) { spc[j] = INFF; path[j] = 0; SC[j] = 0; }
    if (tid == 0) { s_minVal = 0.f; s_i = cur; s_sink = -1; }
    __syncthreads();

    // ---- Dijkstra-like column scan ----
    while (true) {
      int i = s_i;
      float minVal = s_minVal;
      const float* Crow = C + (size_t)i * QN;
      float ui = u_[i];

      float lval = INFF;
      int   lidx = QN;
      for (int j = tid; j < QN; j += 256) {
        if (!SC[j]) {
          float r = minVal + Crow[j] - ui - v_[j];
          if (r < spc[j]) { spc[j] = r; path[j] = i; }
          float s = spc[j];
          if (s < lval) { lval = s; lidx = j; }   // j increasing => first-min ties
        }
      }
      // wave32 butterfly argmin (tie -> lowest column index, matches argmin)
#pragma unroll
      for (int off = 16; off > 0; off >>= 1) {
        float ov = __shfl_xor(lval, off, 32);
        int   oi = __shfl_xor(lidx, off, 32);
        if (ov < lval || (ov == lval && oi < lidx)) { lval = ov; lidx = oi; }
      }
      int wid = tid >> 5;
      if ((tid & 31) == 0) { redV[wid] = lval; redJ[wid] = lidx; }
      __syncthreads();
      if (tid == 0) {
        float bv = redV[0]; int bj = redJ[0];
#pragma unroll
        for (int w = 1; w < 8; ++w)
          if (redV[w] < bv || (redV[w] == bv && redJ[w] < bj)) { bv = redV[w]; bj = redJ[w]; }
        s_minVal = bv;
        SC[bj] = 1;
        int r4 = row4col[bj];
        if (r4 < 0) s_sink = bj; else s_i = r4;
      }
      __syncthreads();
      if (s_sink >= 0) break;
    }

    // ---- dual updates (on pre-augment col4row, like scipy/reference) ----
    float minVal = s_minVal;
    if (tid < TN) {
      if (tid == cur) {
        u_[tid] += minVal;
      } else {
        int c4 = col4row[tid];
        if (c4 >= 0 && SC[c4]) u_[tid] += minVal - spc[c4];
      }
    }
    for (int j = tid; j < QN; j += 256)
      if (SC[j]) v_[j] -= minVal - spc[j];
    __syncthreads();

    // ---- augment alternating path (scalar, thread 0) ----
    if (tid == 0) {
      int j = s_sink;
      while (true) {
        int i = path[j];
        row4col[j] = i;
        int nj = col4row[i];
        col4row[i] = j;
        if (i == cur) break;
        j = nj;
      }
    }
    __syncthreads();
  }

  // ---- argsort by assigned query index via rank-count (all distinct) ----
  if (tid < TN) {
    int c = col4row[tid];
    int rank = 0;
#pragma unroll 4
    for (int s = 0; s < TN; ++s) rank += (col4row[s] < c) ? 1 : 0;
    pred_out[b * TN + rank] = c;    // pred_idx
    tgt_out[b * TN + rank]  = tid;  // tgt_idx
  }
}

// --------------------------------------------------------------------------
extern "C" void kernel_launch(void* const* d_in, const int* in_sizes, int n_in,
                              void* d_out, int out_size, void* d_ws, size_t ws_size,
                              hipStream_t stream) {
  const float* pred_logits = (const float*)d_in[0];   // (B,Q,NC) f32
  const float* pred_coords = (const float*)d_in[1];   // (B,Q,D)  f32
  const int*   tgt_labels  = (const int*)  d_in[2];   // (B,T)    i32
  const float* tgt_joints  = (const float*)d_in[3];   // (B,T,D)  f32

  // workspace: negated softmax probs (5MB) then cost matrix (32MB)
  float* negP = (float*)d_ws;
  float* cost = negP + (size_t)BS * QN * NC;

  // outputs: int32 (pred_idx then tgt_idx), 2*B*T elements, 4B each
  int* out      = (int*)d_out;
  int* pred_out = out;
  int* tgt_out  = out + BS * TN;

  // Stage 1: -softmax over 65536 rows of 20 logits
  softmax_neg_kernel<<<(BS * QN + 255) / 256, 256, 0, stream>>>(pred_logits, negP);

  // Stage 2: 64*8*64 = 32768 WMMA tiles, 8 waves / block -> 4096 blocks
  cost_wmma_kernel<<<(BS * (TN / 16) * (QN / 16)) / 8, 256, 0, stream>>>(
      negP, pred_coords, tgt_labels, tgt_joints, cost);

  // Stage 3: one workgroup per batch
  lsa_kernel<<<BS, 256, 0, stream>>>(cost, pred_out, tgt_out);
}